// GraphAttentionLayer_70471823392981
// MI455X (gfx1250) — compile-verified
//
#include <hip/hip_runtime.h>
#include <math.h>

typedef __attribute__((ext_vector_type(2))) float v2f;
typedef __attribute__((ext_vector_type(8))) float v8f;

#define B_    16
#define N_    1024
#define FIN   1024
#define FOUT  512
#define ROWS  (B_ * N_)          // 16384
#define ALPHA 0.2f
#define NEGINF (-9.0e15f)

static __device__ __forceinline__ v8f wmma_f32(v2f a, v2f b, v8f c) {
  // D = A(16x4,f32) x B(4x16,f32) + C(16x16,f32)  -> v_wmma_f32_16x16x4_f32
  return __builtin_amdgcn_wmma_f32_16x16x4_f32(
      /*neg_a=*/false, a, /*neg_b=*/false, b,
      /*c_mod=*/(short)0, c, /*reuse_a=*/false, /*reuse_b=*/false);
}

// ---------------------------------------------------------------------------
// Kernel 1: h = x @ W   (16384x1024 @ 1024x512), written to hp[:, 512:1024]
// Block: 256 threads (8 waves). Tile: BM=128, BN=64, BK=16.
// Each wave computes a 16x64 strip (4 WMMA accumulators).
// ---------------------------------------------------------------------------
#define BM 128
#define BN 64
#define BK 16

__global__ void gat_gemm_xw(const float* __restrict__ X,
                            const float* __restrict__ W,
                            float* __restrict__ hp) {
  __shared__ float As[BM][BK + 1];
  __shared__ float Bs[BK][BN + 4];
  const int tid = threadIdx.x;
  const int w = tid >> 5, l = tid & 31, hi = l >> 4, lo = l & 15;
  const int m0 = blockIdx.y * BM;
  const int n0 = blockIdx.x * BN;

  v8f acc[4] = {};

  for (int k0 = 0; k0 < FIN; k0 += BK) {
    for (int idx = tid; idx < BM * BK; idx += 256) {
      int r = idx >> 4, c = idx & 15;
      As[r][c] = X[(size_t)(m0 + r) * FIN + k0 + c];
    }
    for (int idx = tid; idx < BK * BN; idx += 256) {
      int r = idx >> 6, c = idx & 63;
      Bs[r][c] = W[(size_t)(k0 + r) * FOUT + n0 + c];
    }
    __syncthreads();
#pragma unroll
    for (int kk = 0; kk < 4; ++kk) {
      v2f a;
      a.x = As[w * 16 + lo][kk * 4 + hi * 2 + 0];
      a.y = As[w * 16 + lo][kk * 4 + hi * 2 + 1];
#pragma unroll
      for (int t = 0; t < 4; ++t) {
        v2f b;
        b.x = Bs[kk * 4 + hi * 2 + 0][t * 16 + lo];
        b.y = Bs[kk * 4 + hi * 2 + 1][t * 16 + lo];
        acc[t] = wmma_f32(a, b, acc[t]);
      }
    }
    __syncthreads();
  }

#pragma unroll
  for (int t = 0; t < 4; ++t)
#pragma unroll
    for (int v = 0; v < 8; ++v) {
      int row = m0 + w * 16 + v + hi * 8;
      int col = n0 + t * 16 + lo;
      hp[(size_t)row * (2 * FOUT) + FOUT + col] = acc[t][v];
    }
}

// ---------------------------------------------------------------------------
// Kernel 2: f1[r] = h[r]·a1, f2[r] = h[r]·a2.  One wave per row.
// ---------------------------------------------------------------------------
__global__ void gat_compute_f(const float* __restrict__ hp,
                              const float* __restrict__ a_vec,
                              float* __restrict__ f1,
                              float* __restrict__ f2) {
  const int wid = threadIdx.x >> 5, lane = threadIdx.x & 31;
  const int row = blockIdx.x * 8 + wid;
  const float* h = hp + (size_t)row * (2 * FOUT) + FOUT;
  float s1 = 0.f, s2 = 0.f;
  for (int c = lane; c < FOUT; c += 32) {
    float v = h[c];
    s1 += v * a_vec[c];
    s2 += v * a_vec[FOUT + c];
  }
#pragma unroll
  for (int off = 16; off > 0; off >>= 1) {
    s1 += __shfl_down(s1, off, 32);
    s2 += __shfl_down(s2, off, 32);
  }
  if (lane == 0) { f1[row] = s1; f2[row] = s2; }
}

// ---------------------------------------------------------------------------
// Kernel 3: fused masked-softmax attention + agg = att @ h (flash style).
// Block: 512 threads (16 waves), one block per (batch, 16-row i-tile).
// Wave w owns output columns [32w, 32w+32). Online softmax over j-tiles of 16.
// agg written to hp[:, 0:512].
// ---------------------------------------------------------------------------
__global__ void gat_attn_agg(const int* __restrict__ adj,
                             const float* __restrict__ f1g,
                             const float* __restrict__ f2g,
                             float* __restrict__ hp) {
  __shared__ float Lt[16][17];     // masked logits tile
  __shared__ float Pt[16][17];     // probabilities tile
  __shared__ float f1s[16], f2s[16], m_run[16], s_run[16], r_s[16];

  const int tid = threadIdx.x;
  const int wid = tid >> 5, l = tid & 31, hi = l >> 4, lo = l & 15;
  const int b = blockIdx.y, i0 = blockIdx.x * 16;
  const int n0w = wid * 32;
  const float* H = hp + (size_t)b * N_ * (2 * FOUT) + FOUT;  // H[j] row stride 1024

  if (tid < 16) {
    f1s[tid]  = f1g[b * N_ + i0 + tid];
    m_run[tid] = -INFINITY;
    s_run[tid] = 0.f;
  }
  v8f acc[2] = {};
  __syncthreads();

  for (int j0 = 0; j0 < N_; j0 += 16) {
    if (tid < 16) f2s[tid] = f2g[b * N_ + j0 + tid];
    __syncthreads();

    if (tid < 256) {
      int i = tid >> 4, jl = tid & 15;
      float e = f1s[i] + f2s[jl];
      e = e > 0.f ? e : ALPHA * e;
      int av = adj[(size_t)(b * N_ + i0 + i) * N_ + j0 + jl];
      Lt[i][jl] = (av > 0) ? e : NEGINF;
    }
    __syncthreads();

    if (tid < 16) {  // per-row online softmax update
      int i = tid;
      float tm = Lt[i][0];
#pragma unroll
      for (int j = 1; j < 16; ++j) tm = fmaxf(tm, Lt[i][j]);
      float mn = fmaxf(m_run[i], tm);
      float rr = expf(m_run[i] - mn);   // first tile: exp(-inf)=0
      float ts = 0.f;
#pragma unroll
      for (int j = 0; j < 16; ++j) {
        float p = expf(Lt[i][j] - mn);
        Pt[i][j] = p;
        ts += p;
      }
      s_run[i] = s_run[i] * rr + ts;
      m_run[i] = mn;
      r_s[i]   = rr;
    }
    __syncthreads();

    // rescale accumulators by per-row factor, then P @ H with WMMA
#pragma unroll
    for (int v = 0; v < 8; ++v) {
      float rv = r_s[v + hi * 8];
      acc[0][v] *= rv;
      acc[1][v] *= rv;
    }
#pragma unroll
    for (int kk = 0; kk < 4; ++kk) {
      v2f a;
      a.x = Pt[lo][kk * 4 + hi * 2 + 0];
      a.y = Pt[lo][kk * 4 + hi * 2 + 1];
      int ja = j0 + kk * 4 + hi * 2;
#pragma unroll
      for (int t = 0; t < 2; ++t) {
        int col = n0w + t * 16 + lo;
        v2f bf;
        bf.x = H[(size_t)(ja + 0) * (2 * FOUT) + col];
        bf.y = H[(size_t)(ja + 1) * (2 * FOUT) + col];
        acc[t] = wmma_f32(a, bf, acc[t]);
      }
    }
    __syncthreads();
  }

#pragma unroll
  for (int t = 0; t < 2; ++t)
#pragma unroll
    for (int v = 0; v < 8; ++v) {
      int row = v + hi * 8;
      float s = s_run[row];
      hp[(size_t)(b * N_ + i0 + row) * (2 * FOUT) + n0w + t * 16 + lo] =
          acc[t][v] / s;
    }
}

// ---------------------------------------------------------------------------
// Kernel 4: out = elu(concat(agg,h) @ fc_w^T + fc_b).  Same WMMA tiling,
// B read transposed from fc_w (512x1024 row-major).
// ---------------------------------------------------------------------------
__global__ void gat_gemm_out(const float* __restrict__ hp,
                             const float* __restrict__ fc_w,
                             const float* __restrict__ fc_b,
                             float* __restrict__ out) {
  __shared__ float As[BM][BK + 1];
  __shared__ float Bs[BK][BN + 4];
  const int tid = threadIdx.x;
  const int w = tid >> 5, l = tid & 31, hi = l >> 4, lo = l & 15;
  const int m0 = blockIdx.y * BM;
  const int n0 = blockIdx.x * BN;

  v8f acc[4] = {};

  for (int k0 = 0; k0 < 2 * FOUT; k0 += BK) {
    for (int idx = tid; idx < BM * BK; idx += 256) {
      int r = idx >> 4, c = idx & 15;
      As[r][c] = hp[(size_t)(m0 + r) * (2 * FOUT) + k0 + c];
    }
    for (int idx = tid; idx < BK * BN; idx += 256) {
      int c = idx >> 4, r = idx & 15;   // contiguous reads along fc_w rows
      Bs[r][c] = fc_w[(size_t)(n0 + c) * (2 * FOUT) + k0 + r];
    }
    __syncthreads();
#pragma unroll
    for (int kk = 0; kk < 4; ++kk) {
      v2f a;
      a.x = As[w * 16 + lo][kk * 4 + hi * 2 + 0];
      a.y = As[w * 16 + lo][kk * 4 + hi * 2 + 1];
#pragma unroll
      for (int t = 0; t < 4; ++t) {
        v2f b;
        b.x = Bs[kk * 4 + hi * 2 + 0][t * 16 + lo];
        b.y = Bs[kk * 4 + hi * 2 + 1][t * 16 + lo];
        acc[t] = wmma_f32(a, b, acc[t]);
      }
    }
    __syncthreads();
  }

#pragma unroll
  for (int t = 0; t < 4; ++t)
#pragma unroll
    for (int v = 0; v < 8; ++v) {
      int row = m0 + w * 16 + v + hi * 8;
      int col = n0 + t * 16 + lo;
      float val = acc[t][v] + fc_b[col];
      val = (val > 0.f) ? val : expm1f(val);   // ELU(alpha=1)
      out[(size_t)row * FOUT + col] = val;
    }
}

// ---------------------------------------------------------------------------
extern "C" void kernel_launch(void* const* d_in, const int* in_sizes, int n_in,
                              void* d_out, int out_size, void* d_ws, size_t ws_size,
                              hipStream_t stream) {
  const float* x    = (const float*)d_in[0];   // (16,1024,1024)
  const int*   adj  = (const int*)  d_in[1];   // (16,1024,1024)
  const float* W    = (const float*)d_in[2];   // (1024,512)
  const float* a    = (const float*)d_in[3];   // (1024,1)
  const float* fc_w = (const float*)d_in[4];   // (512,1024)
  const float* fc_b = (const float*)d_in[5];   // (512,)
  float* out = (float*)d_out;                  // (16,1024,512)

  // Workspace layout: hp (16384 x 1024 concat buffer) | f1 | f2
  float* hp = (float*)d_ws;
  float* f1 = hp + (size_t)ROWS * (2 * FOUT);
  float* f2 = f1 + ROWS;

  // 1) h = x @ W  -> hp[:, 512:1024]
  gat_gemm_xw<<<dim3(FOUT / BN, ROWS / BM), 256, 0, stream>>>(x, W, hp);

  // 2) f1, f2 row dot-products
  gat_compute_f<<<ROWS / 8, 256, 0, stream>>>(hp, a, f1, f2);

  // 3) fused masked softmax + agg -> hp[:, 0:512]
  gat_attn_agg<<<dim3(N_ / 16, B_), 512, 0, stream>>>(adj, f1, f2, hp);

  // 4) out = elu(hp @ fc_w^T + fc_b)
  gat_gemm_out<<<dim3(FOUT / BN, ROWS / BM), 256, 0, stream>>>(hp, fc_w, fc_b, out);
}